// RegionFeatureExtraction_37520834298281
// MI455X (gfx1250) — compile-verified
//
#include <hip/hip_runtime.h>

typedef float v2f __attribute__((ext_vector_type(2)));
typedef float v8f __attribute__((ext_vector_type(8)));
typedef int   v4i __attribute__((vector_size(4 * sizeof(int))));

typedef __attribute__((address_space(1))) v4i* gptr_v4i;   // global
typedef __attribute__((address_space(3))) v4i* lptr_v4i;   // LDS

#define BDIM 256
#define KT   64
#define LSTR 68   // padded LDS row stride (floats): 16B-aligned rows, conflict-free frags
#define NT   49   // 3136 / 64 K-tiles

#if __has_builtin(__builtin_amdgcn_global_load_async_to_lds_b128)
#define HAS_ASYNC_LDS 1
#else
#define HAS_ASYNC_LDS 0
#endif

#if __has_builtin(__builtin_amdgcn_s_wait_asynccnt)
#define WAIT_ASYNC(n) __builtin_amdgcn_s_wait_asynccnt(n)
#elif HAS_ASYNC_LDS
#define WAIT_ASYNC(n) asm volatile("s_wait_asynccnt %0" ::"n"(n) : "memory")
#else
#define WAIT_ASYNC(n) ((void)0)
#endif

// Copy 16 bytes global -> LDS. Async (ASYNCcnt-tracked) when available.
__device__ __forceinline__ void copy16_g2l(const float* __restrict__ gsrc, float* ldst) {
#if HAS_ASYNC_LDS
    // Integer round-trip: global aperture is identity; low 32 bits of a flat
    // LDS address are the LDS offset (ISA aperture rules).
    __builtin_amdgcn_global_load_async_to_lds_b128(
        (gptr_v4i)(unsigned long long)gsrc,
        (lptr_v4i)(unsigned long long)(unsigned int)(unsigned long long)ldst,
        0, 0);
#else
    *(float4*)ldst = *(const float4*)gsrc;
#endif
}

__global__ __launch_bounds__(BDIM)
void region_feats_wmma_kernel(const float* __restrict__ x,
                              const float* __restrict__ pa,
                              const float* __restrict__ cw,
                              const float* __restrict__ cb,
                              float* __restrict__ out)
{
    constexpr int C = 256, P = 32, HW = 3136;

    __shared__ __align__(16) float ldsA[2][P * LSTR];    // raw pa tiles: 2 x 32 x 64
    __shared__ __align__(16) float ldsX[2][64 * LSTR];   // x tiles:      2 x 64 x 64
    __shared__ float attP[4 * KT];                       // partial att sums
    __shared__ float attL[KT];                           // sigmoid att per pixel
    __shared__ float cwL[P];
    __shared__ float biasL;

    const int tid = threadIdx.x;
    const int b   = blockIdx.y;
    const int c0  = blockIdx.x * 64;

    if (tid < P)  cwL[tid] = cw[tid];
    if (tid == 0) biasL = cb[0];

    const float* paB = pa + (size_t)b * P * HW;
    const float* xB  = x  + (size_t)b * C * HW + (size_t)c0 * HW;

    const int lane = tid & 31;
    const int wave = tid >> 5;
    const int wm   = wave & 1;           // M tile 0..1
    const int wn   = wave >> 1;          // N tile 0..3
    const int lrow = lane & 15;
    const int kk   = (lane >> 4) << 1;   // 0 (lanes 0-15) or 2 (lanes 16-31)

    // ---- cooperative staging of one K-tile into buffer `buf` (6 b128 ops/thread) ----
    auto stage = [&](int buf, int hw0) {
        #pragma unroll
        for (int i = 0; i < 2; ++i) {                    // pa: 512 quads
            int e = tid + i * BDIM;
            int p = e >> 4, kq = e & 15;
            copy16_g2l(&paB[p * HW + hw0 + kq * 4], &ldsA[buf][p * LSTR + kq * 4]);
        }
        #pragma unroll
        for (int i = 0; i < 4; ++i) {                    // x: 1024 quads
            int e = tid + i * BDIM;
            int c = e >> 4, kq = e & 15;
            copy16_g2l(&xB[(size_t)c * HW + hw0 + kq * 4], &ldsX[buf][c * LSTR + kq * 4]);
        }
    };

    stage(0, 0);                                          // prologue: tile 0 in flight
    v8f acc = {};

    for (int t = 0; t < NT; ++t) {
        const int cur = t & 1;
        const int hw1 = (t + 1) * KT;

        if (hw1 < HW) {                                   // issue next tile, then wait for
            stage(cur ^ 1, hw1);                          // current one (6 ops in flight)
            WAIT_ASYNC(6);
        } else {
            WAIT_ASYNC(0);
        }
        __syncthreads();                                  // all waves' tile-t data in LDS

        // ---- att partial sums: 256 threads, 8 p-rows each ----
        {
            const int q = tid >> 6, k = tid & 63;
            const float* ar = &ldsA[cur][q * 8 * LSTR + k];
            float s = 0.f;
            #pragma unroll
            for (int p = 0; p < 8; ++p) s += ar[p * LSTR] * cwL[q * 8 + p];
            attP[q * KT + k] = s;
        }
        __syncthreads();
        if (tid < KT) {
            float s = attP[tid] + attP[KT + tid] + attP[2 * KT + tid]
                    + attP[3 * KT + tid] + biasL;
            attL[tid] = 1.0f / (1.0f + __expf(-s));
        }
        __syncthreads();

        // ---- WMMA: D(16x16) += (A*att)(16x4) * B(4x16), att folded into A fragment ----
        const float* aRow = &ldsA[cur][(wm * 16 + lrow) * LSTR];
        const float* bRow = &ldsX[cur][(wn * 16 + lrow) * LSTR];
        #pragma unroll
        for (int k0 = 0; k0 < KT; k0 += 4) {
            v2f a, bb, aw;
            a.x  = aRow[k0 + kk];      a.y  = aRow[k0 + kk + 1];
            aw.x = attL[k0 + kk];      aw.y = attL[k0 + kk + 1];
            bb.x = bRow[k0 + kk];      bb.y = bRow[k0 + kk + 1];
            a *= aw;
            acc = __builtin_amdgcn_wmma_f32_16x16x4_f32(
                false, a, false, bb, (short)0, acc, false, false);
        }
        __syncthreads();   // everyone done reading `cur` before it is restaged at t+1
    }

    // ---- write feats[b, p, c] : out[b*P*C + p*C + c] ----
    // C/D layout: lanes 0-15 -> N=lane, M=v; lanes 16-31 -> N=lane-16, M=v+8
    const int nBase = c0 + wn * 16 + lrow;
    const int mBase = wm * 16 + ((lane >> 4) << 3);
    float* outB = out + (size_t)b * P * C;
    #pragma unroll
    for (int v = 0; v < 8; ++v) {
        outB[(size_t)(mBase + v) * C + nBase] = acc[v];
    }
}

extern "C" void kernel_launch(void* const* d_in, const int* in_sizes, int n_in,
                              void* d_out, int out_size, void* d_ws, size_t ws_size,
                              hipStream_t stream) {
    const float* x  = (const float*)d_in[0];   // [B,C,H,W]
    const float* pa = (const float*)d_in[1];   // [B,P,H,W]
    const float* cw = (const float*)d_in[2];   // [P]
    const float* cb = (const float*)d_in[3];   // [1]
    float* out = (float*)d_out;                // [B, P*C]

    dim3 grid(256 / 64, 32);                   // (C tiles, B)
    region_feats_wmma_kernel<<<grid, BDIM, 0, stream>>>(x, pa, cw, cb, out);
}